// GCNConvWithConstraint_12970801234372
// MI455X (gfx1250) — compile-verified
//
#include <hip/hip_runtime.h>
#include <hip/hip_bf16.h>

typedef __attribute__((ext_vector_type(2))) float v2f;
typedef __attribute__((ext_vector_type(8))) float v8f;

#define IN_CH  128
#define OUT_CH 128

static __device__ __forceinline__ void atomic_add_f32(float* p, float v) {
    __hip_atomic_fetch_add(p, v, __ATOMIC_RELAXED, __HIP_MEMORY_SCOPE_AGENT);
}

// ---------------------------------------------------------------------------
// Phase 0: column renorm of W[out][in], packed directly into WMMA B-fragment
// order:  Wfrag[kg][colTile][lane][2]  (kg = K/4 group, 32 of them;
// colTile = out/16, 8 of them; lane = wave32 lane; 2 consecutive K values).
// Lane mapping for f32 16x16x4 B (4x16):
//   lanes 0-15  hold (K=4kg+0, K=4kg+1) for N=col0+lane
//   lanes 16-31 hold (K=4kg+2, K=4kg+3) for N=col0+lane-16
// One block, 128 threads; thread k owns input channel (column) k.
// ---------------------------------------------------------------------------
__global__ void renorm_pack_kernel(const float* __restrict__ W,
                                   float* __restrict__ Wfrag) {
    int k = threadIdx.x;            // input channel 0..127
    float s = 0.0f;
    #pragma unroll 8
    for (int o = 0; o < OUT_CH; ++o) {
        float w = W[o * IN_CH + k];
        s += w * w;
    }
    float nrm = sqrtf(s);
    float scale = (nrm > 1.0f) ? (1.0f / nrm) : 1.0f;

    const int kg  = k >> 2;
    const int kk  = k & 3;
    const int hi  = (kk >> 1) << 4;   // lane half: 0 or 16
    const int sub = kk & 1;           // which of the 2 packed floats
    #pragma unroll 8
    for (int o = 0; o < OUT_CH; ++o) {
        int cw = o >> 4;
        int m  = o & 15;
        int lane = hi | m;
        Wfrag[(((kg * 8 + cw) * 32) + lane) * 2 + sub] = W[o * IN_CH + k] * scale;
    }
}

// ---------------------------------------------------------------------------
// Phase 1: degree computation. deg initialized to 1.0 (self loop), then +1
// per edge landing on col. Finally deg -> rsqrt(deg) in place.
// ---------------------------------------------------------------------------
__global__ void deg_init_kernel(float* __restrict__ deg, int N) {
    int n = blockIdx.x * blockDim.x + threadIdx.x;
    if (n < N) deg[n] = 1.0f;
}

__global__ void deg_count_kernel(const int* __restrict__ col,
                                 float* __restrict__ deg, int E) {
    int e = blockIdx.x * blockDim.x + threadIdx.x;
    if (e < E) atomic_add_f32(&deg[col[e]], 1.0f);
}

__global__ void deg_rsqrt_kernel(float* __restrict__ deg, int N) {
    int n = blockIdx.x * blockDim.x + threadIdx.x;
    if (n < N) deg[n] = rsqrtf(deg[n]);   // deg >= 1 always (self loop)
}

// ---------------------------------------------------------------------------
// Phase 2: h = x @ Wn^T via V_WMMA_F32_16X16X4_F32.
// Block = 256 threads = 8 waves. Block b owns node rows [32b, 32b+32)
// (TWO 16-row tiles per wave -> shared B fragment, two independent WMMA
// accumulation chains to hide XDL latency). Wave w owns out cols [16w,16w+16).
// h is padded to a 32-row multiple, so stores are unconditional (no exec
// juggling in the epilogue); rows >= N hold clamped duplicates, never read.
// ---------------------------------------------------------------------------
__global__ void gemm_wmma_kernel(const float* __restrict__ x,
                                 const float* __restrict__ Wfrag,
                                 float* __restrict__ h, int N) {
    const int wave  = threadIdx.x >> 5;       // 0..7 -> column tile
    const int lane  = threadIdx.x & 31;
    const int row0  = blockIdx.x * 32;
    const int col0  = wave * 16;
    const int m     = lane & 15;
    const int khalf = (lane >> 4) * 2;        // 0 or 2

    int r0 = row0 + m;        if (r0 >= N) r0 = N - 1;   // clamp keeps EXEC all-1s
    int r1 = row0 + 16 + m;   if (r1 >= N) r1 = N - 1;
    const float* __restrict__ x0 = x + (size_t)r0 * IN_CH + khalf;
    const float* __restrict__ x1 = x + (size_t)r1 * IN_CH + khalf;
    const v2f*   __restrict__ bp = (const v2f*)Wfrag + (wave * 32 + lane);

    v8f c0 = {}, c1 = {};
    #pragma unroll 4
    for (int kg = 0; kg < 32; ++kg) {
        v2f a0 = *(const v2f*)(x0 + kg * 4);
        v2f a1 = *(const v2f*)(x1 + kg * 4);
        v2f b  = bp[kg * (8 * 32)];
        c0 = __builtin_amdgcn_wmma_f32_16x16x4_f32(
                false, a0, false, b, (short)0, c0, false, false);
        c1 = __builtin_amdgcn_wmma_f32_16x16x4_f32(
                false, a1, false, b, (short)0, c1, false, false);
    }

    const int nhalf = (lane >> 4) * 8;
    float* __restrict__ h0 = h + (size_t)(row0 + nhalf) * OUT_CH + col0 + m;
    float* __restrict__ h1 = h0 + (size_t)16 * OUT_CH;
    #pragma unroll
    for (int r = 0; r < 8; ++r) {
        h0[(size_t)r * OUT_CH] = c0[r];
        h1[(size_t)r * OUT_CH] = c1[r];
    }
}

// ---------------------------------------------------------------------------
// Phase 3: out[n] = b + h[n] * dis[n]^2   (bias + self-loop message)
// One thread per (node, 4-channel group).
// ---------------------------------------------------------------------------
__global__ void out_init_kernel(const float* __restrict__ h,
                                const float* __restrict__ dis,
                                const float* __restrict__ bias,
                                float* __restrict__ out, int N) {
    int t = blockIdx.x * blockDim.x + threadIdx.x;
    if (t >= N * 32) return;
    int n = t >> 5;
    int g = t & 31;
    float s = dis[n];
    s = s * s;
    float4 hv = ((const float4*)(h + (size_t)n * OUT_CH))[g];
    float4 bv = ((const float4*)bias)[g];
    float4 o;
    o.x = bv.x + hv.x * s;
    o.y = bv.y + hv.y * s;
    o.z = bv.z + hv.z * s;
    o.w = bv.w + hv.w * s;
    ((float4*)(out + (size_t)n * OUT_CH))[g] = o;
}

// ---------------------------------------------------------------------------
// Phase 4: edge scatter. One wave32 per edge; lane l owns channels 4l..4l+3.
// out[col] += h[row] * dis[row]*dis[col]   via f32 agent-scope atomics.
// h and out are L2-resident (25.6 MB each in a 192 MB L2).
// ---------------------------------------------------------------------------
__global__ void scatter_kernel(const int* __restrict__ rowi,
                               const int* __restrict__ coli,
                               const float* __restrict__ h,
                               const float* __restrict__ dis,
                               float* __restrict__ out, int E) {
    int e = blockIdx.x * (blockDim.x >> 5) + (threadIdx.x >> 5);
    if (e >= E) return;
    int lane = threadIdx.x & 31;
    int r = rowi[e];
    int cdst = coli[e];
    float w = dis[r] * dis[cdst];
    float4 v = ((const float4*)(h + (size_t)r * OUT_CH))[lane];
    float* op = out + (size_t)cdst * OUT_CH + lane * 4;
    atomic_add_f32(op + 0, v.x * w);
    atomic_add_f32(op + 1, v.y * w);
    atomic_add_f32(op + 2, v.z * w);
    atomic_add_f32(op + 3, v.w * w);
}

// ---------------------------------------------------------------------------
// Launch
// ---------------------------------------------------------------------------
extern "C" void kernel_launch(void* const* d_in, const int* in_sizes, int n_in,
                              void* d_out, int out_size, void* d_ws, size_t ws_size,
                              hipStream_t stream) {
    const float* x  = (const float*)d_in[0];   // [N, 128]
    const int*   ei = (const int*)d_in[1];     // [2, E] row-major: row then col
    const float* W  = (const float*)d_in[2];   // [128, 128]
    const float* b  = (const float*)d_in[3];   // [128]
    float* out = (float*)d_out;                // [N, 128]

    const int N = in_sizes[0] / IN_CH;
    const int E = in_sizes[1] / 2;
    const int Npad = (N + 31) & ~31;           // 32-row padding for guard-free stores
    const int* row = ei;
    const int* col = ei + E;

    float* ws    = (float*)d_ws;
    float* Wfrag = ws;                         // 128*128 floats (64 KB)
    float* deg   = ws + IN_CH * OUT_CH;        // Npad floats (-> dis in place)
    float* h     = deg + Npad;                 // Npad*128 floats (~25.6 MB)

    // Phase 0: renorm + pack W into B-fragment order
    renorm_pack_kernel<<<1, 128, 0, stream>>>(W, Wfrag);

    // Phase 1: degrees -> dis = rsqrt(deg)
    deg_init_kernel<<<(N + 255) / 256, 256, 0, stream>>>(deg, N);
    deg_count_kernel<<<(E + 255) / 256, 256, 0, stream>>>(col, deg, E);
    deg_rsqrt_kernel<<<(N + 255) / 256, 256, 0, stream>>>(deg, N);

    // Phase 2: h = x @ Wn^T  (f32 WMMA, 2 row-tiles per wave)
    gemm_wmma_kernel<<<(N + 31) / 32, 256, 0, stream>>>(x, Wfrag, h, N);

    // Phase 3: out = b + h * dis^2  (bias + self loop)
    out_init_kernel<<<(N * 32 + 255) / 256, 256, 0, stream>>>(h, deg, b, out, N);

    // Phase 4: edge scatter with f32 atomics (one wave per edge)
    scatter_kernel<<<(E + 7) / 8, 256, 0, stream>>>(row, col, h, deg, out, E);
}